// Agg_loss_69337952026647
// MI455X (gfx1250) — compile-verified
//
#include <hip/hip_runtime.h>
#include <math.h>

#define B_ 16
#define C_ 4
#define H_ 640
#define W_ 640
#define P_ (H_*W_)
#define N_ 9

typedef float v2f __attribute__((ext_vector_type(2)));
typedef float v8f __attribute__((ext_vector_type(8)));

// ---------------- zero workspace ----------------
__global__ void zero_ws(float* ws, int n) {
  int i = blockIdx.x * blockDim.x + threadIdx.x;
  if (i < n) ws[i] = 0.0f;
}

// ---------------- pass 1: bincounts + per-tag kernel embedding sums ----------------
__global__ __launch_bounds__(256) void count_pass(
    const int* __restrict__ text, const int* __restrict__ kern,
    const int* __restrict__ mask, const float* __restrict__ sv,
    float* __restrict__ ksum, int* __restrict__ kcnt, int* __restrict__ tcnt,
    int* __restrict__ pkm, int* __restrict__ ptm) {
  __shared__ float lks[N_ * C_];
  __shared__ int lkc[N_], ltc[N_], lpk[N_], lpt[N_];
  const int b = blockIdx.y;
  const int tid = threadIdx.x;
  if (tid < N_ * C_) lks[tid] = 0.0f;
  if (tid < N_) { lkc[tid] = 0; ltc[tid] = 0; lpk[tid] = 0; lpt[tid] = 0; }
  __syncthreads();
  const size_t base = (size_t)b * P_;
  const size_t svb  = (size_t)b * C_ * P_;
  for (int p = blockIdx.x * blockDim.x + tid; p < P_; p += gridDim.x * blockDim.x) {
    int k = kern[base + p];
    int t = text[base + p];
    int m = mask[base + p] != 0 ? 1 : 0;
    k = (k < 0) ? 0 : (k >= N_ ? N_ - 1 : k);
    t = (t < 0) ? 0 : (t >= N_ ? N_ - 1 : t);
    atomicAdd(&lkc[k], 1);
    atomicAdd(&ltc[t], 1);
    atomicAdd(&lpk[k * m], 1);   // bincount(kern*mask)
    atomicAdd(&lpt[t * m], 1);   // bincount(text*mask)
#pragma unroll
    for (int c = 0; c < C_; ++c)
      atomicAdd(&lks[k * C_ + c], sv[svb + (size_t)c * P_ + p]);
  }
  __syncthreads();
  if (tid < N_ * C_) atomicAdd(&ksum[b * N_ * C_ + tid], lks[tid]);
  if (tid < N_) {
    atomicAdd(&kcnt[b * N_ + tid], lkc[tid]);
    atomicAdd(&tcnt[b * N_ + tid], ltc[tid]);
    atomicAdd(&pkm [b * N_ + tid], lpk[tid]);
    atomicAdd(&ptm [b * N_ + tid], lpt[tid]);
  }
}

// ---------------- pass 2: kernel means + squared norms ----------------
__global__ void mean_pass(const float* __restrict__ ksum, const int* __restrict__ kcnt,
                          float* __restrict__ kmean, float* __restrict__ knorm) {
  int i = blockIdx.x * blockDim.x + threadIdx.x;
  if (i < B_ * N_) {
    float inv = 1.0f / fmaxf((float)kcnt[i], 1.0f);
    float s = 0.0f;
#pragma unroll
    for (int c = 0; c < C_; ++c) {
      float km = ksum[i * C_ + c] * inv;
      kmean[i * C_ + c] = km;
      s += km * km;
    }
    knorm[i] = s;
  }
}

// ---------------- pass 3: per-pixel loss via WMMA dot(sv, kmean[n]) ----------------
// dist^2[p, t] = ||sv[p]||^2 - 2 * <sv[p], kmean[t]> + ||kmean[t]||^2
// Middle term for all 9 tags at once: [16 pix x 4 ch] x [4 ch x 16 tags] via
// V_WMMA_F32_16X16X4_F32. A layout (ISA 7.12.2, 32-bit A 16x4): lanes 0-15
// carry K=0,1 of pixel M=lane; lanes 16-31 carry K=2,3 of pixel M=lane-16.
__global__ __launch_bounds__(256) void loss_pass(
    const int* __restrict__ text, const float* __restrict__ sv,
    const float* __restrict__ kmean, const float* __restrict__ knorm,
    float* __restrict__ tsum) {
  __shared__ float tile[8][16][16];  // per-wave D tile: dot[pixel][tag]
  __shared__ float lkm[N_ * C_];
  __shared__ float lkn[N_];
  __shared__ float lts[N_];
  const int b    = blockIdx.y;
  const int tid  = threadIdx.x;
  const int lane = tid & 31;
  const int wid  = tid >> 5;
  const int col  = lane & 15;   // pixel row (A) / tag column (B,D)
  const int kh   = lane >> 4;   // K half: channels {0,1} or {2,3}
  if (tid < N_ * C_) lkm[tid] = kmean[b * N_ * C_ + tid];
  if (tid < N_) { lkn[tid] = knorm[b * N_ + tid]; lts[tid] = 0.0f; }
  __syncthreads();

  // B matrix (kmean^T, 4x16): loop-invariant per wave, columns >= 9 are zero
  v2f bm = {0.0f, 0.0f};
  if (col < N_) {
    bm.x = lkm[col * C_ + 2 * kh];
    bm.y = lkm[col * C_ + 2 * kh + 1];
  }

  const size_t svb = (size_t)b * C_ * P_;
  const size_t tb  = (size_t)b * P_;
  const int chunk  = gridDim.x * 128;          // pixels per iteration, all blocks
  const int iters  = (P_ + chunk - 1) / chunk; // uniform trip count => barriers legal

  for (int it = 0; it < iters; ++it) {
    const int wbase = it * chunk + (blockIdx.x * 8 + wid) * 16;
    const int pix   = wbase + col;
    const int pc    = pix < P_ ? pix : P_ - 1;  // clamp: keep EXEC all-1 for WMMA

    const float a0 = sv[svb + (size_t)(2 * kh)     * P_ + pc];
    const float a1 = sv[svb + (size_t)(2 * kh + 1) * P_ + pc];
    // hoisted (clamped) label load: overlaps WMMA + LDS stores
    int t = text[tb + pc];
    t = (t < 0) ? 0 : (t >= N_ ? N_ - 1 : t);
    __builtin_prefetch(&sv[svb + (size_t)(2 * kh) * P_ +
                           (pc + chunk < P_ ? pc + chunk : pc)], 0, 1);

    // ||sv[p]||^2: sum the two channel-halves across the lane xor-16 pair
    float part  = a0 * a0 + a1 * a1;
    float snorm = part + __shfl_xor(part, 16, 32);

    v2f am = {a0, a1};
    v8f acc = {};
    acc = __builtin_amdgcn_wmma_f32_16x16x4_f32(
        false, am, false, bm, (short)0, acc, false, false);

    // D layout: lanes 0-15 hold (M=r, N=lane); lanes 16-31 hold (M=r+8, N=lane-16)
#pragma unroll
    for (int r = 0; r < 8; ++r)
      tile[wid][r + 8 * kh][col] = acc[r];
    __syncthreads();

    if (kh == 0 && pix < P_) {
      float dot   = tile[wid][col][t];
      float d2    = snorm - 2.0f * dot + lkn[t];
      // d2 clamped >= 1e-12 (normal range): raw v_sqrt_f32 is safe
      float dist  = __builtin_amdgcn_sqrtf(fmaxf(d2, 1e-12f));
      float hinge = fmaxf(dist - 0.5f, 0.0f);
      // argument >= 1.0: raw v_log_f32 (log2) * ln2 is safe
      float pl    = __builtin_amdgcn_logf(hinge * hinge + 1.0f) * 0.6931471805599453f;
      atomicAdd(&lts[t], pl);
    }
    __syncthreads();
  }
  if (tid < N_) atomicAdd(&tsum[b * N_ + tid], lts[tid]);
}

// ---------------- pass 4: validity + final scalar ----------------
__global__ void final_pass(const int* __restrict__ pkm, const int* __restrict__ ptm,
                           const int* __restrict__ tcnt, const float* __restrict__ tsum,
                           float* __restrict__ out) {
  __shared__ float sper[B_];
  __shared__ float sbv[B_];
  int b = threadIdx.x;
  if (b < B_) {
    int nk = 0, nt = 0, nv = 0;
    float num = 0.0f;
    for (int n = 1; n < N_; ++n) {   // tag 0 (background) excluded
      bool pk = pkm[b * N_ + n] > 0;
      bool pt = ptm[b * N_ + n] > 0;
      nk += pk; nt += pt;
      if (pk && pt) {
        num += tsum[b * N_ + n] / fmaxf((float)tcnt[b * N_ + n], 1.0f);
        nv++;
      }
    }
    sper[b] = (nv > 0) ? num / (float)nv : 0.0f;
    sbv[b]  = (nk >= 1 && nt >= 1 && nk == nt) ? 1.0f : 0.0f;
  }
  __syncthreads();
  if (b == 0) {
    float s = 0.0f, nb = 0.0f;
    for (int i = 0; i < B_; ++i) { s += sper[i] * sbv[i]; nb += sbv[i]; }
    out[0] = (nb > 0.0f) ? s / nb : 0.0f;
  }
}

extern "C" void kernel_launch(void* const* d_in, const int* in_sizes, int n_in,
                              void* d_out, int out_size, void* d_ws, size_t ws_size,
                              hipStream_t stream) {
  const int*   text = (const int*)d_in[0];
  const int*   kern = (const int*)d_in[1];
  const int*   mask = (const int*)d_in[2];
  const float* sv   = (const float*)d_in[3];
  float* out = (float*)d_out;

  // workspace layout (element units)
  float* ws_f  = (float*)d_ws;
  float* ksum  = ws_f;            // [B][N][C] 576
  float* kmean = ws_f + 576;      // [B][N][C] 576
  float* knorm = ws_f + 1152;     // [B][N]    144
  float* tsum  = ws_f + 1296;     // [B][N]    144
  int*   ws_i  = (int*)d_ws + 1440;
  int*   kcnt  = ws_i;            // [B][N] 144
  int*   tcnt  = ws_i + 144;      // [B][N] 144
  int*   pkm   = ws_i + 288;      // [B][N] 144
  int*   ptm   = ws_i + 432;      // [B][N] 144
  const int WS_ELEMS = 2016;

  zero_ws  <<<(WS_ELEMS + 255) / 256, 256, 0, stream>>>(ws_f, WS_ELEMS);
  count_pass<<<dim3(400, B_), 256, 0, stream>>>(text, kern, mask, sv,
                                                ksum, kcnt, tcnt, pkm, ptm);
  mean_pass <<<1, 160, 0, stream>>>(ksum, kcnt, kmean, knorm);
  loss_pass <<<dim3(320, B_), 256, 0, stream>>>(text, sv, kmean, knorm, tsum);
  final_pass<<<1, 32, 0, stream>>>(pkm, ptm, tcnt, tsum, out);
}